// YonedaUniverse_89498528514849
// MI455X (gfx1250) — compile-verified
//
#include <hip/hip_runtime.h>

// Problem shape (fixed by reference): X (2048,256) f32, probes (512,256) f32,
// out (2048,512) f32 with out[b,p] = mean_d( P[p,d] <= X[b,d] ? 1 : X[b,d] ).
#define DIMK    256
#define TB      64          // batch tile per block
#define TP      64          // probe tile per block
#define DC      32          // dim chunk (double buffered)
#define NCHUNK  (DIMK / DC) // 8
#define ROWPAD  36          // 32 data floats + 4 pad floats per LDS row

typedef int v4i __attribute__((ext_vector_type(4)));
typedef __attribute__((address_space(1))) v4i gbl_v4i;
typedef __attribute__((address_space(3))) v4i lds_v4i;

// ---- CDNA5 async global->LDS DMA (ASYNCcnt path) -------------------------
__device__ __forceinline__ void async_copy_b128(const float* g, float* l) {
#if __has_builtin(__builtin_amdgcn_global_load_async_to_lds_b128)
  __builtin_amdgcn_global_load_async_to_lds_b128(
      (gbl_v4i*)(unsigned long long)g,   // global src (as1), 16B aligned
      (lds_v4i*)l,                       // lds dst (as3), 16B aligned
      0, 0);
#else
  unsigned lds_off = (unsigned)(unsigned long long)(
      __attribute__((address_space(3))) float*)l;
  asm volatile("global_load_async_to_lds_b128 %0, %1, off"
               :: "v"(lds_off), "v"((unsigned long long)g)
               : "memory");
#endif
}

template <int N>
__device__ __forceinline__ void wait_asynccnt() {
#if __has_builtin(__builtin_amdgcn_s_wait_asynccnt)
  __builtin_amdgcn_s_wait_asynccnt(N);
#else
  asm volatile("s_wait_asynccnt %0" :: "i"(N) : "memory");
#endif
}

__device__ __forceinline__ void wg_barrier() {
  asm volatile("" ::: "memory");
  __builtin_amdgcn_s_barrier();
  asm volatile("" ::: "memory");
}

// --------------------------------------------------------------------------
__global__ __launch_bounds__(256) void yoneda_implication_kernel(
    const float* __restrict__ X, const float* __restrict__ P,
    float* __restrict__ out, int NP) {
  // Double-buffered tiles, rows padded to 36 floats (bank-conflict-free b128 reads).
  __shared__ __align__(16) float Xs[2][TB * ROWPAD];
  __shared__ __align__(16) float Ps[2][TP * ROWPAD];

  const int tid = threadIdx.x;
  const int tx  = tid & 15;   // probe group: probes p0 + tx + 16*j
  const int ty  = tid >> 4;   // batch group: batches b0 + 4*ty + i
  const int b0  = blockIdx.y * TB;
  const int p0  = blockIdx.x * TP;

  // Staging layout: 256 threads -> 32 rows x 8 segments of 16B, twice per tile.
  const int lrow = tid >> 3;  // 0..31
  const int lseg = tid & 7;   // 0..7

  auto stage = [&](int buf, int c) {
    const float* gx = X + (size_t)(b0 + lrow) * DIMK + c * DC + lseg * 4;
    async_copy_b128(gx,              &Xs[buf][lrow * ROWPAD + lseg * 4]);
    async_copy_b128(gx + 32 * DIMK,  &Xs[buf][(lrow + 32) * ROWPAD + lseg * 4]);
    const float* gp = P + (size_t)(p0 + lrow) * DIMK + c * DC + lseg * 4;
    async_copy_b128(gp,              &Ps[buf][lrow * ROWPAD + lseg * 4]);
    async_copy_b128(gp + 32 * DIMK,  &Ps[buf][(lrow + 32) * ROWPAD + lseg * 4]);
  };

  float acc[4][4] = {};

  stage(0, 0);  // prologue: chunk 0 -> buffer 0 (4 async ops in flight/wave)

  for (int c = 0; c < NCHUNK; ++c) {
    const int buf = c & 1;
    if (c + 1 < NCHUNK) {
      stage(buf ^ 1, c + 1);   // prefetch next chunk into other buffer
      wait_asynccnt<4>();      // drain chunk c (in-order), keep c+1 in flight
    } else {
      wait_asynccnt<0>();
    }
    wg_barrier();              // chunk c visible to all 8 waves

    const float* xs = &Xs[buf][0];
    const float* ps = &Ps[buf][0];
#pragma unroll
    for (int d = 0; d < DC; d += 4) {
      float4 xv[4], pv[4];
#pragma unroll
      for (int i = 0; i < 4; ++i)
        xv[i] = *(const float4*)&xs[(ty * 4 + i) * ROWPAD + d];
#pragma unroll
      for (int j = 0; j < 4; ++j)
        pv[j] = *(const float4*)&ps[(tx + 16 * j) * ROWPAD + d];
#pragma unroll
      for (int i = 0; i < 4; ++i)
#pragma unroll
        for (int j = 0; j < 4; ++j) {
          acc[i][j] += (pv[j].x <= xv[i].x) ? 1.0f : xv[i].x;
          acc[i][j] += (pv[j].y <= xv[i].y) ? 1.0f : xv[i].y;
          acc[i][j] += (pv[j].z <= xv[i].z) ? 1.0f : xv[i].z;
          acc[i][j] += (pv[j].w <= xv[i].w) ? 1.0f : xv[i].w;
        }
    }
    wg_barrier();              // all waves done reading buf before it refills
  }

  const float inv = 1.0f / (float)DIMK;
#pragma unroll
  for (int i = 0; i < 4; ++i) {
    const size_t rb = (size_t)(b0 + ty * 4 + i) * NP + p0;
#pragma unroll
    for (int j = 0; j < 4; ++j)
      out[rb + tx + 16 * j] = acc[i][j] * inv;   // lanes 0..15 coalesce
  }
}

// --------------------------------------------------------------------------
extern "C" void kernel_launch(void* const* d_in, const int* in_sizes, int n_in,
                              void* d_out, int out_size, void* d_ws, size_t ws_size,
                              hipStream_t stream) {
  (void)n_in; (void)d_ws; (void)ws_size; (void)out_size;
  const float* X = (const float*)d_in[0];
  const float* P = (const float*)d_in[1];
  float* out     = (float*)d_out;

  const int BATCH = in_sizes[0] / DIMK;  // 2048
  const int NP    = in_sizes[1] / DIMK;  // 512

  dim3 grid(NP / TP, BATCH / TB);        // (8, 32) = 256 blocks
  yoneda_implication_kernel<<<grid, 256, 0, stream>>>(X, P, out, NP);
}